// _EPCombineWrapper_70703751627373
// MI455X (gfx1250) — compile-verified
//
#include <hip/hip_runtime.h>

// MoE combine (EP combine) for MI455X / gfx1250.
//
// out = output_buffer; for each selected row r: out[token_idx[r]] += gate[r] * E[r]
//
// Bandwidth-bound (0.35 FLOP/byte) -> no WMMA; the CDNA5 leverage is the memory
// system:
//  - Kernel 1: copy output_buffer -> d_out (output is 128 MB, L2 is 192 MB ->
//    the whole output becomes and stays L2-resident).
//  - Kernel 2: TDM pipeline. One tensor_load_to_lds per tile DMAs 4 rows
//    (32 KB) into LDS (double-buffered, TENSORcnt + s_wait_tensorcnt),
//    waves read staged rows with ds_load_b128 and scatter with no-return
//    global_atomic_add_f32 that resolve in L2. TDM stream uses NT cache
//    policy so it does not evict the L2-resident output.

typedef float        v4f __attribute__((ext_vector_type(4)));
typedef unsigned int v4u __attribute__((ext_vector_type(4)));
typedef int          v4i __attribute__((ext_vector_type(4)));
typedef int          v8i __attribute__((ext_vector_type(8)));

#define THREADS 256
#define ROWS    4        // rows per TDM tile: 4 * 8KB = 32KB; x2 buffers = 64KB LDS

#if __has_builtin(__builtin_amdgcn_tensor_load_to_lds) && \
    __has_builtin(__builtin_amdgcn_s_wait_tensorcnt)
#define USE_TDM 1
#else
#define USE_TDM 0
#endif

// ---------------------------------------------------------------------------
// Kernel 1: output init (copy output_buffer into d_out, lines stay in L2)
// ---------------------------------------------------------------------------
__global__ void init_out_kernel(const v4f* __restrict__ src,
                                v4f* __restrict__ dst, int n4) {
    int i = blockIdx.x * blockDim.x + threadIdx.x;
    if (i < n4) dst[i] = src[i];
}

#if USE_TDM
// ---------------------------------------------------------------------------
// Issue one TDM descriptor: load a [ROWS x d_model] f32 tile into LDS.
// D# group0: count=1 | lds_addr | global_addr[56:0] | type=2
// D# group1: data_size=4B, tensor_dim0=d_model, tensor_dim1=rows_left,
//            tile_dim0=d_model, tile_dim1=ROWS, tensor_dim0_stride=d_model
// ---------------------------------------------------------------------------
__device__ __forceinline__ void tdm_issue_tile(const float* src,
                                               unsigned lds_byte_off,
                                               unsigned d_model,
                                               unsigned rows_left) {
    unsigned long long ga = (unsigned long long)(size_t)src;

    v4u g0;
    g0.x = 1u;                                   // [1:0] count = 1 (valid)
    g0.y = lds_byte_off;                         // [63:32] lds_addr (bytes)
    g0.z = (unsigned)ga;                         // global_addr[31:0]
    g0.w = (unsigned)((ga >> 32) & 0x01FFFFFFull)// global_addr[56:32]
         | (2u << 30);                           // [127:126] type = 2 (image)

    unsigned w0 = (2u << 16);                    // data_size=2 (4B); mask/flags=0
    unsigned w1 = (d_model & 0xFFFFu) << 16;     // tensor_dim0[15:0]; abar=0
    unsigned w2 = ((d_model >> 16) & 0xFFFFu)    // tensor_dim0[31:16]
                | ((rows_left & 0xFFFFu) << 16); // tensor_dim1[15:0]
    unsigned w3 = ((rows_left >> 16) & 0xFFFFu)  // tensor_dim1[31:16]
                | ((d_model & 0xFFFFu) << 16);   // tile_dim0 = d_model
    unsigned w4 = (unsigned)ROWS;                // tile_dim1 = ROWS; tile_dim2=0
    unsigned w5 = d_model;                       // tensor_dim0_stride[31:0]
    unsigned w6 = 0u;                            // stride0 hi | stride1 lo
    unsigned w7 = 0u;                            // stride1 hi

    v8i g1;
    g1[0] = (int)w0; g1[1] = (int)w1; g1[2] = (int)w2; g1[3] = (int)w3;
    g1[4] = (int)w4; g1[5] = (int)w5; g1[6] = (int)w6; g1[7] = (int)w7;

    v4i z4 = {0, 0, 0, 0};                       // groups 2/3 unused (<=2D tile)
#if __clang_major__ >= 23
    v8i z8 = {0, 0, 0, 0, 0, 0, 0, 0};
    __builtin_amdgcn_tensor_load_to_lds(g0, g1, z4, z4, z8, 1 /* cpol: TH_NT */);
#else
    __builtin_amdgcn_tensor_load_to_lds(g0, g1, z4, z4, 1 /* cpol: TH_NT */);
#endif
}

// ---------------------------------------------------------------------------
// Kernel 2 (TDM path): double-buffered DMA pipeline + LDS->global scatter
// ---------------------------------------------------------------------------
__global__ void __launch_bounds__(THREADS)
moe_combine_tdm(const float* __restrict__ expert_outputs,
                const float* __restrict__ gates,
                const int*   __restrict__ token_idx,
                float*       __restrict__ out,
                int num_sel, int d_model) {
    extern __shared__ float smem[];              // 2 * ROWS * d_model floats
    const unsigned lds_base   = (unsigned)(size_t)smem;  // addr[31:0] = LDS offset
    const unsigned tile_bytes = (unsigned)(ROWS * d_model * sizeof(float));
    const int ntiles = num_sel / ROWS;           // 32768/4 = 8192 (divisible)
    const int t      = threadIdx.x;
    const int chunks = d_model / (THREADS * 4);  // = 2 for d_model=2048
    const int row4   = d_model / 4;              // float4 per row

    int tile = blockIdx.x;
    if (tile < ntiles && t < 32) {               // wave 0 issues the DMA
        tdm_issue_tile(expert_outputs + (size_t)tile * ROWS * d_model,
                       lds_base, (unsigned)d_model,
                       (unsigned)(num_sel - tile * ROWS));
    }

    int parity = 0;
    for (; tile < ntiles; tile += gridDim.x) {
        const int  next     = tile + (int)gridDim.x;
        const bool has_next = next < ntiles;

        // Prefetch next tile into the other buffer before draining this one.
        if (has_next && t < 32) {
            tdm_issue_tile(expert_outputs + (size_t)next * ROWS * d_model,
                           lds_base + (unsigned)(parity ^ 1) * tile_bytes,
                           (unsigned)d_model,
                           (unsigned)(num_sel - next * ROWS));
        }
        // TDM ops complete in order: <=1 outstanding => current buffer ready.
        if (has_next) __builtin_amdgcn_s_wait_tensorcnt(1);
        else          __builtin_amdgcn_s_wait_tensorcnt(0);
        __syncthreads();                         // publish buffer to all waves

        const v4f* lbuf = (const v4f*)smem + (size_t)parity * ROWS * row4;
        const int  r0   = tile * ROWS;
        for (int row = 0; row < ROWS; ++row) {
            // Block-uniform -> s_load_b32 (KMcnt path).
            const float g   = gates[r0 + row];
            const int   tok = token_idx[r0 + row];
            float* dst = out + (size_t)(unsigned)tok * (size_t)d_model;
            for (int c = 0; c < chunks; ++c) {
                const int j = t + c * THREADS;
                v4f v = lbuf[row * row4 + j];    // ds_load_b128
                float* d = dst + (size_t)j * 4;
                atomicAdd(d + 0, g * v.x);       // global_atomic_add_f32
                atomicAdd(d + 1, g * v.y);       //   (no-return, resolves in L2)
                atomicAdd(d + 2, g * v.z);
                atomicAdd(d + 3, g * v.w);
            }
        }
        __syncthreads();                         // done reading before reuse
        parity ^= 1;
    }
}
#else
// ---------------------------------------------------------------------------
// Fallback (no TDM builtin): direct NT-load + atomic scatter (round-1 kernel)
// ---------------------------------------------------------------------------
__global__ void __launch_bounds__(THREADS)
moe_combine_direct(const float* __restrict__ expert_outputs,
                   const float* __restrict__ gates,
                   const int*   __restrict__ token_idx,
                   float*       __restrict__ out,
                   int num_sel, int d_model) {
    const int t = threadIdx.x;
    const int iters = d_model / (THREADS * 4);
    for (int r = blockIdx.x; r < num_sel; r += gridDim.x) {
        const float g   = gates[r];
        const int   tok = token_idx[r];
        const v4f* src = (const v4f*)(expert_outputs + (size_t)r * (size_t)d_model);
        float*     dst = out + (size_t)(unsigned)tok * (size_t)d_model;
        int rn = r + gridDim.x;
        if (rn < num_sel) {
            const char* p = (const char*)(expert_outputs +
                                          (size_t)rn * (size_t)d_model) + t * 32;
            __builtin_prefetch(p, 0, 1);
        }
        for (int i = 0; i < iters; ++i) {
            int j = t + i * THREADS;
            v4f v = __builtin_nontemporal_load(&src[j]);
            float* d = dst + (size_t)j * 4;
            atomicAdd(d + 0, g * v.x);
            atomicAdd(d + 1, g * v.y);
            atomicAdd(d + 2, g * v.z);
            atomicAdd(d + 3, g * v.w);
        }
    }
}
#endif

extern "C" void kernel_launch(void* const* d_in, const int* in_sizes, int n_in,
                              void* d_out, int out_size, void* d_ws, size_t ws_size,
                              hipStream_t stream) {
    // setup_inputs() order:
    //   d_in[0] = output_buffer  [num_tokens, d_model] f32 (zeros)
    //   d_in[1] = expert_outputs [num_sel,   d_model] f32
    //   d_in[2] = sorted_gates   [num_sel]            f32
    //   d_in[3] = token_indices  [num_sel]            int
    const float* out_init = (const float*)d_in[0];
    const float* eo       = (const float*)d_in[1];
    const float* gates    = (const float*)d_in[2];
    const int*   idx      = (const int*)d_in[3];
    float*       out      = (float*)d_out;

    const int num_sel = in_sizes[2];                 // 32768
    const int d_model = in_sizes[1] / num_sel;       // 2048

    // 1) Initialize output = output_buffer (d_out is poisoned by the harness).
    const int n4 = out_size / 4;
    init_out_kernel<<<(n4 + THREADS - 1) / THREADS, THREADS, 0, stream>>>(
        (const v4f*)out_init, (v4f*)out, n4);

#if USE_TDM
    // 2) TDM-pipelined scatter-add: 2048 persistent blocks, 4 tiles each,
    //    double-buffered 2 x 32KB LDS staging per block.
    const int    ntiles = num_sel / ROWS;
    const int    blocks = ntiles < 2048 ? ntiles : 2048;
    const size_t shmem  = 2u * ROWS * (size_t)d_model * sizeof(float); // 64 KB
    moe_combine_tdm<<<blocks, THREADS, shmem, stream>>>(
        eo, gates, idx, out, num_sel, d_model);
#else
    moe_combine_direct<<<4096, THREADS, 0, stream>>>(
        eo, gates, idx, out, num_sel, d_model);
#endif
}